// EnsemblePooling_23063974379776
// MI455X (gfx1250) — compile-verified
//
#include <hip/hip_runtime.h>
#include <math.h>
#include <stdint.h>

#define NUM_GRAPHS 1024
#define HID 256
#define ROWS 16            // rows per tile
#define NW 8               // waves per block (wave32)
#define BLOCK 256
#define ROWBYTES (HID * 4)               // 1024 B per row
#define TILEBYTES (ROWS * ROWBYTES)      // 16 KB per tile
#define NLOADS (TILEBYTES / (BLOCK * 16))  // 4 async b128 per thread per tile

#if defined(__has_builtin)
#if __has_builtin(__builtin_amdgcn_global_load_async_to_lds_b128)
#define ASYNC_BUILTIN 1
#endif
#if __has_builtin(__builtin_amdgcn_s_wait_asynccnt)
#define WAIT_BUILTIN 1
#endif
#endif

typedef int v4i __attribute__((ext_vector_type(4)));
typedef __attribute__((address_space(1))) v4i gv4i;  // global (__device__) int4
typedef __attribute__((address_space(3))) v4i lv4i;  // LDS (__shared__) int4

// Copy 16 bytes from (uniform global base + divergent byte offset) into LDS
// via the gfx1250 async-to-LDS path (tracked by ASYNCcnt).
__device__ __forceinline__ void async_cp16(const char* gbase_uniform,
                                           uint32_t off, char* lds_dst) {
#if defined(ASYNC_BUILTIN)
  // Signature (from clang diagnostic): (int4 __device__*, int4 __shared__*, Ii, Ii)
  __builtin_amdgcn_global_load_async_to_lds_b128(
      (gv4i*)(gbase_uniform + off), (lv4i*)lds_dst, 0, 0);
#else
  uint32_t l = (uint32_t)(uintptr_t)lds_dst;
  asm volatile("global_load_async_to_lds_b128 %0, %1, %2"
               :
               : "v"(l), "v"(off), "s"(gbase_uniform)
               : "memory");
#endif
}

__device__ __forceinline__ void wait_async_prev() {  // keep newest tile in flight
#if defined(WAIT_BUILTIN)
  __builtin_amdgcn_s_wait_asynccnt(NLOADS);
#else
  asm volatile("s_wait_asynccnt 4" ::: "memory");
#endif
}

__device__ __forceinline__ void wait_async_all() {
#if defined(WAIT_BUILTIN)
  __builtin_amdgcn_s_wait_asynccnt(0);
#else
  asm volatile("s_wait_asynccnt 0" ::: "memory");
#endif
}

// Wave-cooperative 32-ary lower_bound on the sorted batch array.
// Each round: 32 parallel probes + ballot -> 32x range narrowing.
// Invariants: all idx < lo have batch[idx] < key; (hi == n) or batch[hi] >= key.
__device__ __forceinline__ int lower_bound_wave(const int* __restrict__ batch,
                                                int n, int key, int lane) {
  int lo = 0, hi = n;
  while (hi > lo) {
    const int len = hi - lo;
    const int stride = (len + 31) >> 5;  // ceil(len/32)
    const int p = lo + lane * stride;
    const bool pred = (p < hi) && (batch[p] < key);
    const unsigned long long m = __ballot(pred);  // wave32: bits [31:0]
    const int c = __popcll(m);  // prefix length of "value < key" probes
    if (c == 0) break;          // probe at lo says batch[lo] >= key -> ans = lo
    const int pc = lo + c * stride;
    lo = lo + (c - 1) * stride + 1;           // batch[p_{c-1}] < key
    hi = (c < 32 && pc < hi) ? pc : hi;       // batch[p_c] >= key (if sampled)
  }
  return lo;
}

__global__ __launch_bounds__(BLOCK) void ensemble_pool_kernel(
    const float* __restrict__ x, const int* __restrict__ batch,
    const float* __restrict__ att_w, const float* __restrict__ att_b,
    float* __restrict__ out, int n) {
  __shared__ float tiles[2][ROWS * HID];   // 32 KB double buffer
  __shared__ float red[3][NW][HID];        // 24 KB cross-wave reduction
  __shared__ int seg[2];

  const int g = (int)blockIdx.x;
  const int tid = (int)threadIdx.x;
  const int lane = tid & 31;
  const int wav = tid >> 5;

  // Segment bounds: wave 0 -> start(g), wave 1 -> start(g+1). ~4 dependent loads.
  if (wav < 2) {
    const int r = lower_bound_wave(batch, n, g + wav, lane);
    if (lane == 0) seg[wav] = r;
  }
  __syncthreads();
  const int start = seg[0];
  const int end = seg[1];
  const int R = end - start;
  const int T = (R + ROWS - 1) / ROWS;

  // Per-lane slices of the attention weight: channels [4l,4l+3] and [128+4l,131+4l].
  const float4 w0 = *(const float4*)(att_w + 4 * lane);
  const float4 w1 = *(const float4*)(att_w + 128 + 4 * lane);
  const float bias = att_b[0];

  float4 s0 = make_float4(0.f, 0.f, 0.f, 0.f), s1 = s0;   // segment sum
  float4 a0 = s0, a1 = s0;                                 // attention sum
  float4 m0 = make_float4(-INFINITY, -INFINITY, -INFINITY, -INFINITY), m1 = m0;

  const char* xb = (const char*)x;

  // Prologue: issue tile 0.
  if (T > 0) {
    const int rem0 = min(ROWS, R);
    const uint32_t maxoff = (uint32_t)rem0 * ROWBYTES - 16u;
    const char* gb = xb + (size_t)start * ROWBYTES;
    char* lb = (char*)&tiles[0][0];
#pragma unroll
    for (int p = 0; p < NLOADS; ++p) {
      uint32_t off = (uint32_t)(p * (BLOCK * 16) + tid * 16);
      if (off > maxoff) off = maxoff;  // tail-clamp: exactly NLOADS issues/wave
      async_cp16(gb, off, lb + off);
    }
  }

  for (int t = 0; t < T; ++t) {
    const bool more = (t + 1 < T);
    if (more) {
      // Prefetch next tile into the other buffer (safe: last read 2 tiles ago,
      // protected by trailing barrier), then wait only for the *previous* tile.
      const int rb = start + (t + 1) * ROWS;
      const int rem = min(ROWS, end - rb);
      const uint32_t maxoff = (uint32_t)rem * ROWBYTES - 16u;
      const char* gb = xb + (size_t)rb * ROWBYTES;
      char* lb = (char*)&tiles[(t + 1) & 1][0];
#pragma unroll
      for (int p = 0; p < NLOADS; ++p) {
        uint32_t off = (uint32_t)(p * (BLOCK * 16) + tid * 16);
        if (off > maxoff) off = maxoff;
        async_cp16(gb, off, lb + off);
      }
      wait_async_prev();
    } else {
      wait_async_all();
    }
    __syncthreads();  // every wave waited for its own loads -> tile t complete

    const float* buf = &tiles[t & 1][0];
    const int rem = min(ROWS, end - (start + t * ROWS));
#pragma unroll
    for (int q = 0; q < 2; ++q) {
      const int lr = wav + q * NW;  // wave-uniform row selection
      if (lr < rem) {
        const float* row = buf + lr * HID;
        const float4 v0 = *(const float4*)(row + 4 * lane);
        const float4 v1 = *(const float4*)(row + 128 + 4 * lane);
        float part = v0.x * w0.x + v0.y * w0.y + v0.z * w0.z + v0.w * w0.w +
                     v1.x * w1.x + v1.y * w1.y + v1.z * w1.z + v1.w * w1.w;
#pragma unroll
        for (int o = 16; o >= 1; o >>= 1) part += __shfl_xor(part, o, 32);
        const float wt = 1.0f / (1.0f + expf(-(part + bias)));
        s0.x += v0.x; s0.y += v0.y; s0.z += v0.z; s0.w += v0.w;
        s1.x += v1.x; s1.y += v1.y; s1.z += v1.z; s1.w += v1.w;
        m0.x = fmaxf(m0.x, v0.x); m0.y = fmaxf(m0.y, v0.y);
        m0.z = fmaxf(m0.z, v0.z); m0.w = fmaxf(m0.w, v0.w);
        m1.x = fmaxf(m1.x, v1.x); m1.y = fmaxf(m1.y, v1.y);
        m1.z = fmaxf(m1.z, v1.z); m1.w = fmaxf(m1.w, v1.w);
        a0.x = fmaf(wt, v0.x, a0.x); a0.y = fmaf(wt, v0.y, a0.y);
        a0.z = fmaf(wt, v0.z, a0.z); a0.w = fmaf(wt, v0.w, a0.w);
        a1.x = fmaf(wt, v1.x, a1.x); a1.y = fmaf(wt, v1.y, a1.y);
        a1.z = fmaf(wt, v1.z, a1.z); a1.w = fmaf(wt, v1.w, a1.w);
      }
    }
    __syncthreads();  // done reading buf[t&1] before it is refilled at t+1
  }

  // Cross-wave reduction through LDS (conflict-free b128 stores/loads).
  *(float4*)&red[0][wav][4 * lane] = s0;
  *(float4*)&red[0][wav][128 + 4 * lane] = s1;
  *(float4*)&red[1][wav][4 * lane] = m0;
  *(float4*)&red[1][wav][128 + 4 * lane] = m1;
  *(float4*)&red[2][wav][4 * lane] = a0;
  *(float4*)&red[2][wav][128 + 4 * lane] = a1;
  __syncthreads();

  const int c = tid;  // one channel per thread
  float ssum = 0.f, smax = -INFINITY, satt = 0.f;
#pragma unroll
  for (int w = 0; w < NW; ++w) {
    ssum += red[0][w][c];
    smax = fmaxf(smax, red[1][w][c]);
    satt += red[2][w][c];
  }
  float* og = out + (size_t)g * (3 * HID);
  og[c] = ssum / fmaxf((float)R, 1.0f);  // mean (empty -> 0, like counts>=1 clamp)
  og[HID + c] = smax;                    // max  (empty -> -inf, segment_max identity)
  og[2 * HID + c] = satt;                // attention-weighted sum
}

extern "C" void kernel_launch(void* const* d_in, const int* in_sizes, int n_in,
                              void* d_out, int out_size, void* d_ws,
                              size_t ws_size, hipStream_t stream) {
  (void)n_in; (void)out_size; (void)d_ws; (void)ws_size;
  const float* x = (const float*)d_in[0];
  const int* batch = (const int*)d_in[1];       // integer input -> const int*
  const float* att_w = (const float*)d_in[2];
  const float* att_b = (const float*)d_in[3];
  float* out = (float*)d_out;
  const int n = in_sizes[1];                    // number of nodes
  ensemble_pool_kernel<<<dim3(NUM_GRAPHS), dim3(BLOCK), 0, stream>>>(
      x, batch, att_w, att_b, out, n);
}